// OfficialSS2DOperator_15393162788909
// MI455X (gfx1250) — compile-verified
//
#include <hip/hip_runtime.h>
#include <math.h>

typedef float v2f __attribute__((ext_vector_type(2)));
typedef float v8f __attribute__((ext_vector_type(8)));

#define NTOK 8192      // B * H * W
#define LSEQ 1024
#define DIN  512
#define XLD  48        // DT_RANK + 2*D_STATE

__device__ __forceinline__ float silu_f(float x) { return x / (1.f + expf(-x)); }

// ---------------------------------------------------------------------------
// fp32 WMMA GEMM:  C(MxN) = alpha * A(MxK) * W(NxK)^T  [+ epilogues]
// mode 0: store (+bias if non-null)
// mode 1: accumulate  C += alpha*AB
// mode 2: C = softplus(AB + bias)
// mode 3: C[b, n, h, w] = AB + bias   (scatter token-major -> BCHW)
// One wave computes a 32 x (16*JT) tile. Requirements (host guarantees):
//   M % 32 == 0, N % (16*JT) == 0, K % 4 == 0, K >= 8.
// Inner loop is guard-free: no EXEC masking around WMMA; one-step software
// pipeline (prefetch k+4 fragments before issuing k's WMMAs).
// ---------------------------------------------------------------------------
template<int JT>
__global__ __launch_bounds__(256) void gemm_f32_wmma(
    const float* __restrict__ A, int lda,
    const float* __restrict__ W, int ldw,
    float* __restrict__ C, int ldc,
    int M, int N, int K,
    const float* __restrict__ bias,
    float alpha, int mode)
{
    const int lane  = threadIdx.x & 31;
    const int wave  = threadIdx.x >> 5;
    const int tilesN = N / (16 * JT);
    const int tilesM = M >> 5;
    const int gw = blockIdx.x * 8 + wave;
    if (gw >= tilesM * tilesN) return;
    const int tm = (gw / tilesN) * 32;
    const int tn = (gw % tilesN) * (16 * JT);
    const int half = lane >> 4;                  // 0 -> K{0,1}, 1 -> K{2,3}
    const int mr   = lane & 15;

    v8f zero = {};
    v8f acc[2][JT];
#pragma unroll
    for (int i = 0; i < 2; ++i)
#pragma unroll
        for (int j = 0; j < JT; ++j) acc[i][j] = zero;

    const float* Ar0 = A + (long)(tm + mr) * lda + 2 * half;
    const float* Ar1 = A + (long)(tm + 16 + mr) * lda + 2 * half;
    const float* Wr[JT];
#pragma unroll
    for (int j = 0; j < JT; ++j)
        Wr[j] = W + (long)(tn + j * 16 + mr) * ldw + 2 * half;

    // prologue: load fragments for k = 0
    v2f a0, a1, bb[JT];
    a0.x = Ar0[0]; a0.y = Ar0[1];
    a1.x = Ar1[0]; a1.y = Ar1[1];
#pragma unroll
    for (int j = 0; j < JT; ++j) { bb[j].x = Wr[j][0]; bb[j].y = Wr[j][1]; }

    for (int k0 = 0; k0 < K - 4; k0 += 4) {
        const int nk = k0 + 4;
        // prefetch next k-step fragments
        v2f na0, na1, nb[JT];
        na0.x = Ar0[nk]; na0.y = Ar0[nk + 1];
        na1.x = Ar1[nk]; na1.y = Ar1[nk + 1];
#pragma unroll
        for (int j = 0; j < JT; ++j) { nb[j].x = Wr[j][nk]; nb[j].y = Wr[j][nk + 1]; }
        // compute on current fragments
#pragma unroll
        for (int j = 0; j < JT; ++j) {
            acc[0][j] = __builtin_amdgcn_wmma_f32_16x16x4_f32(
                false, a0, false, bb[j], (short)0, acc[0][j], false, false);
            acc[1][j] = __builtin_amdgcn_wmma_f32_16x16x4_f32(
                false, a1, false, bb[j], (short)0, acc[1][j], false, false);
        }
        a0 = na0; a1 = na1;
#pragma unroll
        for (int j = 0; j < JT; ++j) bb[j] = nb[j];
    }
    // epilogue k-step
#pragma unroll
    for (int j = 0; j < JT; ++j) {
        acc[0][j] = __builtin_amdgcn_wmma_f32_16x16x4_f32(
            false, a0, false, bb[j], (short)0, acc[0][j], false, false);
        acc[1][j] = __builtin_amdgcn_wmma_f32_16x16x4_f32(
            false, a1, false, bb[j], (short)0, acc[1][j], false, false);
    }

#pragma unroll
    for (int i = 0; i < 2; ++i) {
#pragma unroll
        for (int j = 0; j < JT; ++j) {
            const int ncol = tn + j * 16 + mr;
#pragma unroll
            for (int r = 0; r < 8; ++r) {
                const int row = tm + i * 16 + half * 8 + r;
                float v = acc[i][j][r] * alpha;
                if (mode == 0) {
                    if (bias) v += bias[ncol];
                    C[(long)row * ldc + ncol] = v;
                } else if (mode == 1) {
                    C[(long)row * ldc + ncol] += v;
                } else if (mode == 2) {
                    v += bias[ncol];
                    C[(long)row * ldc + ncol] = (v > 20.f) ? v : log1pf(expf(v));
                } else { // mode 3: BCHW scatter
                    v += bias[ncol];
                    const int b = row >> 10;
                    const int l = row & 1023;
                    C[(((long)b * 256 + ncol) << 10) + l] = v;
                }
            }
        }
    }
}

// x (B,256,32,32) -> xrow (8192,256) token-major (h*W+w) and xcol (w*H+h)
__global__ void transpose_tok(const float* __restrict__ x,
                              float* __restrict__ xrow,
                              float* __restrict__ xcol)
{
    const int c = threadIdx.x;          // 0..255
    const int tok = blockIdx.x;         // 0..8191
    const int b = tok >> 10;
    const int hw = tok & 1023;
    const int h = hw >> 5, w = hw & 31;
    const float v = x[(((long)b * 256 + c) << 10) + hw];
    xrow[(long)tok * 256 + c] = v;
    xcol[((((long)b) << 10) + (w << 5) + h) * 256 + c] = v;
}

// causal depthwise conv1d (k=4, left pad 3) + SiLU, in scan order
__global__ void conv_silu(const float* __restrict__ xz,   // (8,1024,1024)
                          const float* __restrict__ cw,   // (512,4)
                          const float* __restrict__ cb,   // (512)
                          float* __restrict__ ucv,        // (8,1024,512)
                          int rev)
{
    const long idx = (long)blockIdx.x * blockDim.x + threadIdx.x;
    if (idx >= (long)NTOK * DIN) return;
    const int d = (int)(idx & 511);
    const int t = (int)((idx >> 9) & 1023);
    const int b = (int)(idx >> 19);
    float acc = cb[d];
#pragma unroll
    for (int j = 0; j < 4; ++j) {
        const int tt = t - 3 + j;
        if (tt >= 0) {
            const int lm = rev ? (1023 - tt) : tt;
            acc += cw[d * 4 + j] * xz[((((long)b) << 10) + lm) * 1024 + d];
        }
    }
    ucv[idx] = silu_f(acc);
}

// Selective scan: block = 16 states x 16 channels; per-thread chain over L.
__global__ __launch_bounds__(256) void scan_kernel(
    const float* __restrict__ dt,    // (8,1024,512) softplus'd
    const float* __restrict__ ucv,   // (8,1024,512)
    const float* __restrict__ xdbl,  // (8,1024,48): [.,16:32)=B, [.,32:48)=C
    const float* __restrict__ xz,    // (8,1024,1024) for z gate
    const float* __restrict__ A_log, // (512,16)
    const float* __restrict__ Dsk,   // (512)
    float* __restrict__ ysum,        // (8,1024,512) in original token order
    int dir, int accum)
{
    const int n  = threadIdx.x & 15;
    const int d  = blockIdx.x * 16 + (threadIdx.x >> 4);
    const int b  = blockIdx.y;
    const float Aval = -expf(A_log[d * 16 + n]);
    const float Dv   = Dsk[d];
    const long base512  = ((long)b << 10) * 512;
    const long base48   = ((long)b << 10) * 48;
    const long base1024 = ((long)b << 10) * 1024;
    float h = 0.f;
    for (int t = 0; t < 1024; ++t) {
        const float dtv = dt[base512 + (long)t * 512 + d];
        const float uv  = ucv[base512 + (long)t * 512 + d];
        const float Bn  = xdbl[base48 + (long)t * 48 + 16 + n];
        const float Cn  = xdbl[base48 + (long)t * 48 + 32 + n];
        h = expf(dtv * Aval) * h + (dtv * uv) * Bn;
        float p = h * Cn;
        p += __shfl_xor(p, 8, 16);
        p += __shfl_xor(p, 4, 16);
        p += __shfl_xor(p, 2, 16);
        p += __shfl_xor(p, 1, 16);
        if (n == 0) {
            const int ts = (dir & 1) ? (1023 - t) : t;   // index into pair's xz
            int ltok;
            if (dir < 2) ltok = ts;
            else         ltok = ((ts & 31) << 5) + (ts >> 5); // (w*H+h)->(h*W+w)
            const float zv = xz[base1024 + (long)ts * 1024 + 512 + d];
            const float y = (p + uv * Dv) * silu_f(zv);
            const long oi = base512 + (long)ltok * 512 + d;
            if (accum) ysum[oi] += y; else ysum[oi] = y;
        }
    }
}

// depthwise 3x3 conv (zero pad) + exact GELU, token-major in/out
__global__ void dwconv_gelu(const float* __restrict__ xrow, // (8192,256)
                            const float* __restrict__ dww,  // (256,9)
                            float* __restrict__ out)        // (8192,256)
{
    const int c = threadIdx.x;
    const int tok = blockIdx.x;
    const int b = tok >> 10;
    const int hw = tok & 1023;
    const int h = hw >> 5, w = hw & 31;
    float acc = 0.f;
#pragma unroll
    for (int dy = 0; dy < 3; ++dy) {
        const int hh = h + dy - 1;
        if (hh < 0 || hh > 31) continue;
#pragma unroll
        for (int dx = 0; dx < 3; ++dx) {
            const int ww2 = w + dx - 1;
            if (ww2 < 0 || ww2 > 31) continue;
            acc += dww[c * 9 + dy * 3 + dx] *
                   xrow[((((long)b) << 10) + (hh << 5) + ww2) * 256 + c];
        }
    }
    out[(long)tok * 256 + c] = 0.5f * acc * (1.f + erff(acc * 0.7071067811865475f));
}

// LayerNorm over 256 channels, one wave per token (8 channels per lane)
__global__ __launch_bounds__(256) void layernorm_kernel(
    const float* __restrict__ m, const float* __restrict__ g,
    const float* __restrict__ bt, float* __restrict__ out)
{
    const int lane = threadIdx.x & 31;
    const int wave = threadIdx.x >> 5;
    const int tok = blockIdx.x * 8 + wave;
    const float* row = m + (long)tok * 256;
    float vals[8];
    float s = 0.f;
#pragma unroll
    for (int i = 0; i < 8; ++i) { vals[i] = row[lane + 32 * i]; s += vals[i]; }
#pragma unroll
    for (int k = 16; k >= 1; k >>= 1) s += __shfl_xor(s, k, 32);
    const float mu = s * (1.f / 256.f);
    float vs = 0.f;
#pragma unroll
    for (int i = 0; i < 8; ++i) { const float dv = vals[i] - mu; vs += dv * dv; }
#pragma unroll
    for (int k = 16; k >= 1; k >>= 1) vs += __shfl_xor(vs, k, 32);
    const float inv = rsqrtf(vs * (1.f / 256.f) + 1e-5f);
    float* orow = out + (long)tok * 256;
#pragma unroll
    for (int i = 0; i < 8; ++i) {
        const int c = lane + 32 * i;
        orow[c] = (vals[i] - mu) * inv * g[c] + bt[c];
    }
}

static void launch_gemm(const float* A, int lda, const float* W, int ldw,
                        float* C, int ldc, int M, int N, int K,
                        const float* bias, float alpha, int mode, hipStream_t s)
{
    if ((N & 63) == 0) {
        const int tiles = (M / 32) * (N / 64);
        gemm_f32_wmma<4><<<(tiles + 7) / 8, 256, 0, s>>>(
            A, lda, W, ldw, C, ldc, M, N, K, bias, alpha, mode);
    } else { // N % 48 == 0 path (N = 48)
        const int tiles = (M / 32) * (N / 48);
        gemm_f32_wmma<3><<<(tiles + 7) / 8, 256, 0, s>>>(
            A, lda, W, ldw, C, ldc, M, N, K, bias, alpha, mode);
    }
}

extern "C" void kernel_launch(void* const* d_in, const int* in_sizes, int n_in,
                              void* d_out, int out_size, void* d_ws, size_t ws_size,
                              hipStream_t stream)
{
    (void)in_sizes; (void)n_in; (void)out_size; (void)ws_size;
    const float* x      = (const float*)d_in[0];
    const float* W_in[2]    = { (const float*)d_in[1],  (const float*)d_in[10] };
    const float* conv_w[2]  = { (const float*)d_in[2],  (const float*)d_in[11] };
    const float* conv_b[2]  = { (const float*)d_in[3],  (const float*)d_in[12] };
    const float* W_xproj[2] = { (const float*)d_in[4],  (const float*)d_in[13] };
    const float* W_dt[2]    = { (const float*)d_in[5],  (const float*)d_in[14] };
    const float* b_dt[2]    = { (const float*)d_in[6],  (const float*)d_in[15] };
    const float* A_log[2]   = { (const float*)d_in[7],  (const float*)d_in[16] };
    const float* Dsk[2]     = { (const float*)d_in[8],  (const float*)d_in[17] };
    const float* W_out[2]   = { (const float*)d_in[9],  (const float*)d_in[18] };
    const float* dw_w   = (const float*)d_in[19];
    const float* pw_w   = (const float*)d_in[20];
    const float* pw_b   = (const float*)d_in[21];
    const float* ln_g   = (const float*)d_in[22];
    const float* ln_b   = (const float*)d_in[23];
    const float* proj_w = (const float*)d_in[24];
    const float* proj_b = (const float*)d_in[25];

    float* ws   = (float*)d_ws;
    float* xrow = ws;                          // 8192*256
    float* xcol = xrow + (long)NTOK * 256;     // 8192*256
    float* xzr  = xcol + (long)NTOK * 256;     // 8192*1024
    float* xzc  = xzr  + (long)NTOK * 1024;    // 8192*1024
    float* ucv  = xzc  + (long)NTOK * 1024;    // 8192*512  (reused per dir)
    float* xdbl = ucv  + (long)NTOK * 512;     // 8192*48   (reused per dir)
    float* dtb  = xdbl + (long)NTOK * XLD;     // 8192*512  (reused per dir)
    float* ysum[2];
    ysum[0] = dtb + (long)NTOK * 512;          // 8192*512  (lr+rl)
    ysum[1] = ysum[0] + (long)NTOK * 512;      // 8192*512  (tb+bt)
    float* gel  = ysum[1] + (long)NTOK * 512;  // 8192*256
    float* mix  = gel + (long)NTOK * 256;      // 8192*256
    float* lnb  = mix + (long)NTOK * 256;      // 8192*256

    // 1) token-major transposes
    transpose_tok<<<NTOK, 256, 0, stream>>>(x, xrow, xcol);

    // 2) input projections (shared within each direction pair)
    launch_gemm(xrow, 256, W_in[0], 256, xzr, 1024, NTOK, 1024, 256,
                nullptr, 1.f, 0, stream);
    launch_gemm(xcol, 256, W_in[1], 256, xzc, 1024, NTOK, 1024, 256,
                nullptr, 1.f, 0, stream);

    // 3) four directional scans (sequential, shared scratch)
    for (int dir = 0; dir < 4; ++dir) {
        const int pair = dir >> 1, rev = dir & 1;
        const float* xz = pair ? xzc : xzr;
        conv_silu<<<((long)NTOK * DIN) / 256, 256, 0, stream>>>(
            xz, conv_w[pair], conv_b[pair], ucv, rev);
        launch_gemm(ucv, 512, W_xproj[pair], 512, xdbl, XLD, NTOK, XLD, 512,
                    nullptr, 1.f, 0, stream);
        launch_gemm(xdbl, XLD, W_dt[pair], 16, dtb, 512, NTOK, 512, 16,
                    b_dt[pair], 1.f, 2, stream);
        scan_kernel<<<dim3(32, 8), 256, 0, stream>>>(
            dtb, ucv, xdbl, xz, A_log[pair], Dsk[pair], ysum[pair], dir, rev);
    }

    // 4) local branch: dw3x3 + GELU, then pointwise GEMM (+bias) into mixed
    dwconv_gelu<<<NTOK, 256, 0, stream>>>(xrow, dw_w, gel);
    launch_gemm(gel, 256, pw_w, 256, mix, 256, NTOK, 256, 256,
                pw_b, 1.f, 0, stream);

    // 5) out-projections accumulate: mixed += 0.25 * (ysum_pair @ W_out^T)
    launch_gemm(ysum[0], 512, W_out[0], 512, mix, 256, NTOK, 256, 512,
                nullptr, 0.25f, 1, stream);
    launch_gemm(ysum[1], 512, W_out[1], 512, mix, 256, NTOK, 256, 512,
                nullptr, 0.25f, 1, stream);

    // 6) LayerNorm + final projection with fused BCHW scatter
    layernorm_kernel<<<NTOK / 8, 256, 0, stream>>>(mix, ln_g, ln_b, lnb);
    launch_gemm(lnb, 256, proj_w, 256, (float*)d_out, 256, NTOK, 256, 256,
                proj_b, 1.f, 3, stream);
}